// NRBS_29824252903580
// MI455X (gfx1250) — compile-verified
//
#include <hip/hip_runtime.h>
#include <hip/hip_bf16.h>
#include <math.h>

#define N_FULL 50000
#define NLAT   10
#define MU     32
#define BATCH  8
#define WAVES  4
#define GP     120   // partial-reduction blocks for encode

typedef __attribute__((ext_vector_type(2))) float v2f;
typedef __attribute__((ext_vector_type(8))) float v8f;

// ---- packed f32 VOP3P helpers (CDNA5) -------------------------------------
__device__ __forceinline__ v2f pk_fma_f32(v2f a, v2f b, v2f c) {
  v2f d;
  asm("v_pk_fma_f32 %0, %1, %2, %3" : "=v"(d) : "v"(a), "v"(b), "v"(c));
  return d;
}
__device__ __forceinline__ v2f pk_add_f32(v2f a, v2f b) {
  v2f d;
  asm("v_pk_add_f32 %0, %1, %2" : "=v"(d) : "v"(a), "v"(b));
  return d;
}

// ---- kernel 1: split-K partial sums of encoded = x @ enc_w.T --------------
__global__ __launch_bounds__(256)
void nrbs_encode_partial(const float* __restrict__ x,
                         const float* __restrict__ enc_w,
                         float* __restrict__ part) {
  __shared__ float sred[BATCH * NLAT];
  const int tid = threadIdx.x;
  if (tid < BATCH * NLAT) sred[tid] = 0.f;
  __syncthreads();

  float acc[BATCH * NLAT];
#pragma unroll
  for (int j = 0; j < BATCH * NLAT; ++j) acc[j] = 0.f;

  for (int p = blockIdx.x * blockDim.x + tid; p < N_FULL;
       p += gridDim.x * blockDim.x) {
    float xv[BATCH];
#pragma unroll
    for (int b = 0; b < BATCH; ++b) xv[b] = x[(size_t)b * N_FULL + p];
    float wv[NLAT];
#pragma unroll
    for (int i = 0; i < NLAT; ++i) wv[i] = enc_w[(size_t)i * N_FULL + p];
#pragma unroll
    for (int b = 0; b < BATCH; ++b)
#pragma unroll
      for (int i = 0; i < NLAT; ++i)
        acc[b * NLAT + i] = fmaf(xv[b], wv[i], acc[b * NLAT + i]);
  }
  // wave32 butterfly reduce, then one LDS float-atomic per wave
#pragma unroll
  for (int j = 0; j < BATCH * NLAT; ++j)
    for (int off = 16; off > 0; off >>= 1)
      acc[j] += __shfl_xor(acc[j], off, 32);
  if ((tid & 31) == 0)
    for (int j = 0; j < BATCH * NLAT; ++j) atomicAdd(&sred[j], acc[j]);
  __syncthreads();
  if (tid < BATCH * NLAT) part[blockIdx.x * (BATCH * NLAT) + tid] = sred[tid];
}

// ---- kernel 2: fold partials + bias -> enc[8][10] -------------------------
__global__ __launch_bounds__(128)
void nrbs_encode_reduce(const float* __restrict__ part,
                        const float* __restrict__ enc_b,
                        float* __restrict__ enc, int nparts) {
  const int t = threadIdx.x;
  if (t < BATCH * NLAT) {
    float s = enc_b[t % NLAT];
    for (int g = 0; g < nparts; ++g) s += part[g * (BATCH * NLAT) + t];
    enc[t] = s;
  }
}

// ---- kernel 3: fused bandwidth-GEMM (WMMA) + window + smooth + output -----
__global__ __launch_bounds__(WAVES * 32)
void nrbs_main(const float* __restrict__ enc,      // [8][10]
               const float* __restrict__ decoder,  // [10][N]
               const float* __restrict__ bw,       // [10][10][N]
               const int*   __restrict__ nbrs,     // [N][32]
               float* __restrict__ out)            // [8][N]
{
  __shared__ float lds_inv2[WAVES][NLAT][32][8];   // per-wave private regions

  const int lane = threadIdx.x & 31;
  const int wv   = threadIdx.x >> 5;
  const int tile = blockIdx.x * WAVES + wv;
  const int p0   = tile * 32;
  if (p0 >= N_FULL) return;   // wave-uniform

  const int m16   = lane & 15;
  const int khalf = lane >> 4;

  // A = encoded [M=b x K=k], ISA 16x4 f32 layout: VGPR0 -> K=kb+2*khalf,
  // VGPR1 -> K=kb+2*khalf+1 ; zero-padded for m>=8 / k>=10.  Because A holds
  // the zeros, B may carry garbage in those K slots (products vanish), so B
  // loads below are UNCONDITIONAL with clamped addresses -> no exec branches.
  v2f A[3];
#pragma unroll
  for (int t = 0; t < 3; ++t) {
    const int k0 = t * 4 + 2 * khalf;
    A[t].x = (m16 < BATCH && k0     < NLAT) ? enc[m16 * NLAT + k0]     : 0.f;
    A[t].y = (m16 < BATCH && k0 + 1 < NLAT) ? enc[m16 * NLAT + k0 + 1] : 0.f;
  }

  // i-invariant element offsets for the six B loads of each half-tile:
  // column p = clamp(p0 + ht*16 + m16), row k = clamp(t*4 + 2*khalf (+1))
  int boff[2][3][2];
#pragma unroll
  for (int ht = 0; ht < 2; ++ht) {
    const int pcl = min(p0 + ht * 16 + m16, N_FULL - 1);
#pragma unroll
    for (int t = 0; t < 3; ++t) {
      const int kx = min(t * 4 + 2 * khalf,     NLAT - 1);
      const int ky = min(t * 4 + 2 * khalf + 1, NLAT - 1);
      boff[ht][t][0] = kx * N_FULL + pcl;
      boff[ht][t][1] = ky * N_FULL + pcl;
    }
  }

  // ---- phase 1: w_pre[b, p] = sum_k enc[b,k]*bw[i,k,p] via WMMA -----------
#pragma unroll 1
  for (int i = 0; i < NLAT; ++i) {
    const float* __restrict__ bwi = bw + (size_t)i * NLAT * N_FULL;
#pragma unroll
    for (int ht = 0; ht < 2; ++ht) {
      v8f c = {};
#pragma unroll
      for (int t = 0; t < 3; ++t) {
        v2f Bv;
        Bv.x = bwi[boff[ht][t][0]];
        Bv.y = bwi[boff[ht][t][1]];
        c = __builtin_amdgcn_wmma_f32_16x16x4_f32(false, A[t], false, Bv,
                                                  (short)0, c, false, false);
      }
      // lanes 0..15: c[v] = w_pre[b=v, p] for 16 consecutive p.
      // sigmoid folded: inv2 = 1/(w*32)^2 = (1+exp(-x))^2 / 1024
      if (lane < 16) {
        const int pl = ht * 16 + lane;
#pragma unroll
        for (int v = 0; v < 8; ++v) {
          const float t1 = 1.f + __expf(-c[v]);
          lds_inv2[wv][i][pl][v] = t1 * t1 * (1.f / 1024.f);
        }
      }
    }
  }
  // per-wave LDS only: compiler-inserted s_wait_dscnt orders write->read.

  // ---- phase 2: window + normalize + smooth + final contraction -----------
  const int  p  = p0 + lane;
  const bool pv = p < N_FULL;
  const int  pc = pv ? p : (N_FULL - 1);
  const int* __restrict__ nrow = nbrs + (size_t)pc * MU;

  float acc_out[BATCH] = {0.f, 0.f, 0.f, 0.f, 0.f, 0.f, 0.f, 0.f};

#pragma unroll 1
  for (int i = 0; i < NLAT; ++i) {
    v2f iv[4];
    const v2f* ivp = (const v2f*)&lds_inv2[wv][i][lane][0];
#pragma unroll
    for (int q = 0; q < 4; ++q) iv[q] = ivp[q];

    v2f accw[4], accg[4];
#pragma unroll
    for (int q = 0; q < 4; ++q) {
      accw[q].x = 0.f; accw[q].y = 0.f;
      accg[q].x = 0.f; accg[q].y = 0.f;
    }
    const float* __restrict__ drow = decoder + (size_t)i * N_FULL;

#pragma unroll 2
    for (int mc = 0; mc < MU; mc += 4) {
      const int4 nv = *(const int4*)(nrow + mc);
      const int nvv[4] = {nv.x, nv.y, nv.z, nv.w};
#pragma unroll
      for (int mm = 0; mm < 4; ++mm) {
        const int   m  = mc + mm;
        const float g  = drow[nvv[mm]];
        const float m2 = (float)(m * m);
        v2f nm2;  nm2.x = -m2; nm2.y = -m2;
        v2f g2;   g2.x  = g;   g2.y  = g;
        v2f one2; one2.x = 1.f; one2.y = 1.f;
#pragma unroll
        for (int q = 0; q < 4; ++q) {
          v2f win = pk_fma_f32(nm2, iv[q], one2);   // 1 - m^2 * inv2
          win.x = fmaxf(win.x, 0.f);
          win.y = fmaxf(win.y, 0.f);
          accw[q] = pk_add_f32(accw[q], win);       // window mass
          accg[q] = pk_fma_f32(g2, win, accg[q]);   // g-weighted sum
        }
      }
    }
    // out[b] += enc[b,i] * (accg/accw)   (accw >= 1 since m=0 term == 1)
#pragma unroll
    for (int q = 0; q < 4; ++q) {
      const float e0 = enc[(2 * q + 0) * NLAT + i];
      const float e1 = enc[(2 * q + 1) * NLAT + i];
      acc_out[2 * q + 0] = fmaf(e0 * accg[q].x, __builtin_amdgcn_rcpf(accw[q].x),
                                acc_out[2 * q + 0]);
      acc_out[2 * q + 1] = fmaf(e1 * accg[q].y, __builtin_amdgcn_rcpf(accw[q].y),
                                acc_out[2 * q + 1]);
    }
  }

  if (pv) {
#pragma unroll
    for (int b = 0; b < BATCH; ++b) out[(size_t)b * N_FULL + p] = acc_out[b];
  }
}

// ---------------------------------------------------------------------------
extern "C" void kernel_launch(void* const* d_in, const int* in_sizes, int n_in,
                              void* d_out, int out_size, void* d_ws, size_t ws_size,
                              hipStream_t stream) {
  const float* x       = (const float*)d_in[0];  // [8][50000]
  const float* enc_w   = (const float*)d_in[1];  // [10][50000]
  const float* enc_b   = (const float*)d_in[2];  // [10]
  const float* decoder = (const float*)d_in[3];  // [10][50000]
  const float* bwl     = (const float*)d_in[4];  // [10][10][50000]
  const int*   nbrs    = (const int*)d_in[5];    // [50000][32]
  float*       out     = (float*)d_out;          // [8][50000]

  float* ws_part = (float*)d_ws;                 // [GP][80]
  float* ws_enc  = ws_part + GP * BATCH * NLAT;  // [80]

  nrbs_encode_partial<<<GP, 256, 0, stream>>>(x, enc_w, ws_part);
  nrbs_encode_reduce<<<1, 128, 0, stream>>>(ws_part, enc_b, ws_enc, GP);

  const int tiles  = (N_FULL + 31) / 32;
  const int blocks = (tiles + WAVES - 1) / WAVES;
  nrbs_main<<<blocks, WAVES * 32, 0, stream>>>(ws_enc, decoder, bwl, nbrs, out);
}